// LowRankModl_6073083756563
// MI455X (gfx1250) — compile-verified
//
#include <hip/hip_runtime.h>
#include <math.h>
#include <stdint.h>

typedef float v2f __attribute__((ext_vector_type(2)));
typedef float v8f __attribute__((ext_vector_type(8)));

#define Bb 2
#define Tt 12
#define Cc 10
#define Hh 256
#define Ww 256
#define Sv 3
#define HW 65536            // Hh*Ww
#define NBT (Bb*Tt*HW)      // 1572864
#define NBSV (Bb*Sv*HW)     // 393216
#define LAM 0.05f
#define TWO_PI 6.28318530717958647692f
#define TSTRIDE 18          // LDS tile row stride (float2), keeps 16B units aligned

__device__ __forceinline__ v8f wmma4(v2f a, v2f b, v8f c) {
    // D(16x16,f32) = A(16x4,f32) x B(4x16,f32) + C
    return __builtin_amdgcn_wmma_f32_16x16x4_f32(false, a, false, b, (short)0, c, false, false);
}

// ---------------------------------------------------------------------------
// Per-wave 256-point DFT with shift-in/shift-out (G[y] = shift(F(shift(x)))),
// ortho scale 1/16, exponent sign `sgn` (+1 = ifft, -1 = fft).
// n = n2 + 16*n1, k = k1 + 16*k2:
//   A[n2][k1]   = sum_n1 x[n2+16*n1] * W16^{n1*k1}     (WMMA matmul 1)
//   Bm[n2][k1]  = A * W256^{n2*k1}                     (twiddle)
//   X[k1+16*k2] = sum_n2 W16^{k2*n2} * Bm[n2][k1]      (WMMA matmul 2)
// Shifts folded as index^128. xin/xout may alias; scr = per-wave 256-float2
// LDS scratch; tw = 256-entry LDS table of e^{+2*pi*i*j/256}.
// ---------------------------------------------------------------------------
__device__ void fft256_wave(const float2* xin, int sx,
                            float2* xout, int so,
                            float2* scr, const float2* tw, float sgn)
{
    const int lane = threadIdx.x & 31;
    const int m    = lane & 15;
    const int hi   = lane >> 4;
    const int koff = hi << 1;

    v2f ar[4], ai[4], nai[4];
#pragma unroll
    for (int q = 0; q < 4; ++q) {
        int k0 = 4*q + koff;
        float2 e0 = xin[((m + 16*k0)      ^ 128) * sx];
        float2 e1 = xin[((m + 16*k0 + 16) ^ 128) * sx];
        ar[q][0] = e0.x;  ar[q][1] = e1.x;
        ai[q][0] = e0.y;  ai[q][1] = e1.y;
        nai[q][0] = -e0.y; nai[q][1] = -e1.y;
    }
    v8f cr = {}; v8f ci = {};
#pragma unroll
    for (int q = 0; q < 4; ++q) {
        int kb = 4*q + koff;
        float2 w0 = tw[((kb * m) & 15) << 4];
        float2 w1 = tw[(((kb + 1) * m) & 15) << 4];
        v2f br; br[0] = w0.x;       br[1] = w1.x;
        v2f bi; bi[0] = sgn * w0.y; bi[1] = sgn * w1.y;
        cr = wmma4(ar[q],  br, cr);
        cr = wmma4(nai[q], bi, cr);
        ci = wmma4(ar[q],  bi, ci);
        ci = wmma4(ai[q],  br, ci);
    }
#pragma unroll
    for (int r8 = 0; r8 < 8; ++r8) {
        int n2 = r8 + 8*hi;
        float2 w = tw[(n2 * m) & 255];
        float wr = w.x, wi = sgn * w.y;
        scr[n2*16 + m] = make_float2(cr[r8]*wr - ci[r8]*wi,
                                     cr[r8]*wi + ci[r8]*wr);
    }
    v8f dr = {}; v8f di = {};
#pragma unroll
    for (int q = 0; q < 4; ++q) {
        int ka = 4*q + koff;
        float2 w0 = tw[((ka * m) & 15) << 4];
        float2 w1 = tw[(((ka + 1) * m) & 15) << 4];
        v2f Ar;  Ar[0] = w0.x;        Ar[1] = w1.x;
        v2f Ai2; Ai2[0] = sgn * w0.y; Ai2[1] = sgn * w1.y;
        v2f nAi; nAi[0] = -Ai2[0];    nAi[1] = -Ai2[1];
        float2 b0 = scr[ka*16 + m];
        float2 b1 = scr[(ka + 1)*16 + m];
        v2f Br; Br[0] = b0.x; Br[1] = b1.x;
        v2f Bi; Bi[0] = b0.y; Bi[1] = b1.y;
        dr = wmma4(Ar,  Br, dr);
        dr = wmma4(nAi, Bi, dr);
        di = wmma4(Ar,  Bi, di);
        di = wmma4(Ai2, Br, di);
    }
#pragma unroll
    for (int r8 = 0; r8 < 8; ++r8) {
        int k2 = r8 + 8*hi;
        int oi = (m + 16*k2) ^ 128;
        xout[oi * so] = make_float2(dr[r8] * 0.0625f, di[r8] * 0.0625f);
    }
}

// ---------------------------------------------------------------------------
// Row pass of R(p): img = S[b,c] * sum_s L[b,t,s] p[b,s], then G+ along w.
// Covers coils [c0, c0+nc). tmp layout: [bt*nc + ci][h][w].
// ---------------------------------------------------------------------------
__global__ __launch_bounds__(256) void lrm_rowfwd(const float2* __restrict__ pv,
        const float* __restrict__ tbr, const float* __restrict__ tbi,
        const float* __restrict__ sr, const float* __restrict__ si,
        int c0, int nc, float2* __restrict__ tmp)
{
    __shared__ float2 tw[256];
    __shared__ float2 wbuf[8][256];
    __shared__ float2 wscr[8][256];
    int tid = threadIdx.x;
    { float s_, c_; __sincosf(TWO_PI * (float)tid * (1.0f/256.0f), &s_, &c_);
      tw[tid] = make_float2(c_, s_); }
    __syncthreads();

    int wv = tid >> 5, lane = tid & 31;
    int row = blockIdx.x * 8 + wv;       // [0, B*T*nc*H)
    int h = row & 255, btc = row >> 8;
    int ci = btc % nc, bt = btc / nc;
    int b = bt / Tt, t = bt - b * Tt;
    int c = c0 + ci;

    float lr[Sv], li[Sv];
#pragma unroll
    for (int s = 0; s < Sv; ++s) { int k = (b*Tt + t)*Sv + s; lr[s] = tbr[k]; li[s] = tbi[k]; }

    const float2* p0 = pv + ((size_t)(b*Sv)*Hh + h) * Ww;
    int sbase = ((b*Cc + c)*Hh + h) * Ww;
    for (int i = lane; i < 256; i += 32) {
        float ex = 0.f, ey = 0.f;
#pragma unroll
        for (int s = 0; s < Sv; ++s) {
            float2 pq = p0[(size_t)s * HW + i];
            ex += lr[s]*pq.x - li[s]*pq.y;
            ey += lr[s]*pq.y + li[s]*pq.x;
        }
        float sxr = sr[sbase + i], sxi = si[sbase + i];
        wbuf[wv][i] = make_float2(sxr*ex - sxi*ey, sxr*ey + sxi*ex);
    }
    fft256_wave(wbuf[wv], 1, tmp + (size_t)btc*HW + (size_t)h*Ww, 1, wscr[wv], tw, 1.0f);
}

// ---------------------------------------------------------------------------
// Column pass, in-place on tmp[btc][h][w]. One 256x16 tile per block, 16 waves.
// Tile staged to LDS with CDNA5 async global->LDS B128 copies.
// mode 0: single G- along h (adjoint of Y).
// mode 1: G+ along h, * mask, G- along h (middle of R(p)).
// ---------------------------------------------------------------------------
__global__ __launch_bounds__(512) void lrm_col(float2* __restrict__ tmp,
        const float* __restrict__ mask, int nc, int mode)
{
    __shared__ float2 tw[256];
    __shared__ __align__(16) float2 tile[256 * TSTRIDE];
    __shared__ float2 wbuf[16][256];
    __shared__ float2 wscr[16][256];
    int tid = threadIdx.x;
    if (tid < 256) { float s_, c_; __sincosf(TWO_PI * (float)tid * (1.0f/256.0f), &s_, &c_);
                     tw[tid] = make_float2(c_, s_); }
    int btc = blockIdx.x >> 4;
    int bt = btc / nc;
    int w0 = (blockIdx.x & 15) * 16;
    size_t base = (size_t)btc * HW;
    size_t mbase = (size_t)bt * HW;

    // ---- async global -> LDS tile copy (4 x 16B per thread) ----
    {
        unsigned ldsbase = (unsigned)(uintptr_t)(&tile[0]);
        const float2* gsrc = tmp + base + w0;
#pragma unroll
        for (int i = 0; i < 4; ++i) {
            int lin = tid + i * 512;            // 16-byte unit index [0,2048)
            int row = lin >> 3;                 // [0,256)
            int cp  = lin & 7;                  // column pair [0,8)
            unsigned lofs = ldsbase + (unsigned)((row * TSTRIDE + cp * 2) * 8);
            unsigned long long ga =
                (unsigned long long)(uintptr_t)(gsrc + (size_t)row * Ww + cp * 2);
            asm volatile("global_load_async_to_lds_b128 %0, %1, off"
                         :: "v"(lofs), "v"(ga) : "memory");
        }
        asm volatile("s_wait_asynccnt 0x0" ::: "memory");
    }
    __syncthreads();

    int wv = tid >> 5, lane = tid & 31;          // wave wv owns tile column wv
    if (mode == 0) {
        fft256_wave(tile + wv, TSTRIDE, tile + wv, TSTRIDE, wscr[wv], tw, -1.0f);
    } else {
        fft256_wave(tile + wv, TSTRIDE, wbuf[wv], 1, wscr[wv], tw, 1.0f);
        for (int i = lane; i < 256; i += 32) {
            float mk = mask[mbase + (size_t)i*Ww + w0 + wv];
            float2 v = wbuf[wv][i];
            wbuf[wv][i] = make_float2(v.x * mk, v.y * mk);
        }
        fft256_wave(wbuf[wv], 1, tile + wv, TSTRIDE, wscr[wv], tw, -1.0f);
    }
    __syncthreads();

    int lcol = tid & 15, row0 = tid >> 4;        // row0 in [0,32)
#pragma unroll
    for (int i = 0; i < 8; ++i) {
        int rr = row0 + 32*i;
        tmp[base + (size_t)rr*Ww + w0 + lcol] = tile[rr*TSTRIDE + lcol];
    }
}

// ---------------------------------------------------------------------------
// Row adjoint pass: for each coil in [c0,c0+nc): G- along w, * conj(S[b,c]),
// accumulated in registers; one write (or RMW if acc) of comb[b,t,h,w].
// ---------------------------------------------------------------------------
__global__ __launch_bounds__(256) void lrm_rowadj(const float2* __restrict__ tmp,
        const float* __restrict__ sr, const float* __restrict__ si,
        int c0, int nc, float2* __restrict__ comb, int acc)
{
    __shared__ float2 tw[256];
    __shared__ float2 wbuf[8][256];
    __shared__ float2 wscr[8][256];
    int tid = threadIdx.x;
    { float s_, c_; __sincosf(TWO_PI * (float)tid * (1.0f/256.0f), &s_, &c_);
      tw[tid] = make_float2(c_, s_); }
    __syncthreads();

    int wv = tid >> 5, lane = tid & 31;
    int row = blockIdx.x * 8 + wv;       // [0, B*T*H)
    int h = row & 255, bt = row >> 8;
    int b = bt / Tt;

    float2 racc[8];
#pragma unroll
    for (int k = 0; k < 8; ++k) racc[k] = make_float2(0.f, 0.f);

    for (int ci = 0; ci < nc; ++ci) {
        size_t tbase = (size_t)(bt*nc + ci)*HW + (size_t)h*Ww;
        for (int i = lane; i < 256; i += 32) wbuf[wv][i] = tmp[tbase + i];
        fft256_wave(wbuf[wv], 1, wbuf[wv], 1, wscr[wv], tw, -1.0f);
        int sbase = ((b*Cc + c0 + ci)*Hh + h) * Ww;
#pragma unroll
        for (int k = 0; k < 8; ++k) {
            int i = lane + 32*k;
            float2 v = wbuf[wv][i];
            float sxr = sr[sbase + i], sxi = si[sbase + i];
            racc[k].x += sxr*v.x + sxi*v.y;     // conj(S)*v
            racc[k].y += sxr*v.y - sxi*v.x;
        }
    }
    size_t obase = (size_t)bt*HW + (size_t)h*Ww;
#pragma unroll
    for (int k = 0; k < 8; ++k) {
        int i = lane + 32*k;
        if (acc) { float2 o = comb[obase + i];
                   comb[obase + i] = make_float2(o.x + racc[k].x, o.y + racc[k].y); }
        else       comb[obase + i] = racc[k];
    }
}

// Pack coils [c0,c0+nc) of Y (separate re/im planes) into tmp float2.
__global__ __launch_bounds__(256) void lrm_pack(const float* __restrict__ kr,
        const float* __restrict__ ki, int c0, int nc, float2* __restrict__ tmp)
{
    int idx = blockIdx.x * 256 + threadIdx.x;     // [0, NBT*nc)
    int btc = idx >> 16, rem = idx & (HW - 1);
    int ci = btc % nc, bt = btc / nc;
    size_t src = ((size_t)bt * Cc + c0 + ci) * HW + rem;
    tmp[idx] = make_float2(kr[src], ki[src]);
}

// Ap = sum_t conj(L) * comb + LAM * p
__global__ __launch_bounds__(256) void lrm_proj_ap(const float2* __restrict__ comb,
        const float* __restrict__ tbr, const float* __restrict__ tbi,
        const float2* __restrict__ pv, float2* __restrict__ Ap)
{
    int idx = blockIdx.x * 256 + threadIdx.x;     // [0, NBSV)
    int pix = idx & (HW - 1);
    int bs = idx >> 16;
    int s = bs % Sv, b = bs / Sv;
    float ax = 0.f, ay = 0.f;
#pragma unroll
    for (int t = 0; t < Tt; ++t) {
        int li = (b*Tt + t)*Sv + s;
        float Lr = tbr[li], Li = tbi[li];
        float2 cv = comb[(size_t)(b*Tt + t)*HW + pix];
        ax += Lr*cv.x + Li*cv.y;
        ay += Lr*cv.y - Li*cv.x;
    }
    float2 pq = pv[idx];
    Ap[idx] = make_float2(ax + LAM*pq.x, ay + LAM*pq.y);
}

// b_vec = sum_t conj(L)*comb + LAM*MO ; r = p = b_vec ; z = 0
__global__ __launch_bounds__(256) void lrm_proj_init(const float2* __restrict__ comb,
        const float* __restrict__ tbr, const float* __restrict__ tbi,
        const float* __restrict__ mor, const float* __restrict__ moi,
        float2* __restrict__ rv, float2* __restrict__ pv, float2* __restrict__ zv)
{
    int idx = blockIdx.x * 256 + threadIdx.x;
    int pix = idx & (HW - 1);
    int bs = idx >> 16;
    int s = bs % Sv, b = bs / Sv;
    float ax = 0.f, ay = 0.f;
#pragma unroll
    for (int t = 0; t < Tt; ++t) {
        int li = (b*Tt + t)*Sv + s;
        float Lr = tbr[li], Li = tbi[li];
        float2 cv = comb[(size_t)(b*Tt + t)*HW + pix];
        ax += Lr*cv.x + Li*cv.y;
        ay += Lr*cv.y - Li*cv.x;
    }
    float2 bvv = make_float2(ax + LAM*mor[idx], ay + LAM*moi[idx]);
    rv[idx] = bvv; pv[idx] = bvv; zv[idx] = make_float2(0.f, 0.f);
}

// partial of vdot(a,b) = sum conj(a)*b ; 512 elements per block
__global__ __launch_bounds__(256) void lrm_dot(const float2* __restrict__ a,
        const float2* __restrict__ b, float2* __restrict__ part)
{
    int tid = threadIdx.x;
    float sx = 0.f, sy = 0.f;
#pragma unroll
    for (int j = 0; j < 2; ++j) {
        int i = blockIdx.x * 512 + j * 256 + tid;
        float2 av = a[i], bv = b[i];
        sx += av.x*bv.x + av.y*bv.y;
        sy += av.x*bv.y - av.y*bv.x;
    }
    for (int off = 16; off > 0; off >>= 1) {
        sx += __shfl_down(sx, off, 32);
        sy += __shfl_down(sy, off, 32);
    }
    __shared__ float2 red[8];
    if ((tid & 31) == 0) red[tid >> 5] = make_float2(sx, sy);
    __syncthreads();
    if (tid == 0) {
        float rx = 0.f, ry = 0.f;
        for (int k = 0; k < 8; ++k) { rx += red[k].x; ry += red[k].y; }
        part[blockIdx.x] = make_float2(rx, ry);
    }
}

__global__ __launch_bounds__(256) void lrm_dot_final(const float2* __restrict__ part,
        float* __restrict__ scal)
{
    int tid = threadIdx.x;
    float sx = 0.f, sy = 0.f;
#pragma unroll
    for (int j = 0; j < 3; ++j) { float2 v = part[tid + j*256]; sx += v.x; sy += v.y; }
    for (int off = 16; off > 0; off >>= 1) {
        sx += __shfl_down(sx, off, 32);
        sy += __shfl_down(sy, off, 32);
    }
    __shared__ float2 red[8];
    if ((tid & 31) == 0) red[tid >> 5] = make_float2(sx, sy);
    __syncthreads();
    if (tid == 0) {
        float rx = 0.f, ry = 0.f;
        for (int k = 0; k < 8; ++k) { rx += red[k].x; ry += red[k].y; }
        scal[8] = rx; scal[9] = ry;
    }
}

// scal: [0,1]=rs_old [2,3]=alpha [4,5]=beta [6]=done [8,9]=last dot result
__global__ void lrm_s_init(float* scal) { scal[0] = scal[8]; scal[1] = scal[9]; scal[6] = 0.f; }

__global__ void lrm_s_alpha(float* scal) {
    if (scal[6] > 0.5f) { scal[2] = 0.f; scal[3] = 0.f; return; }
    float dx = scal[8] + 1e-20f, dy = scal[9];
    float den = dx*dx + dy*dy;
    float rx = scal[0], ry = scal[1];
    scal[2] = (rx*dx + ry*dy) / den;
    scal[3] = (ry*dx - rx*dy) / den;
}

__global__ void lrm_s_beta(float* scal) {
    if (scal[6] > 0.5f) return;
    float nx = scal[8], ny = scal[9];
    float dx = scal[0] + 1e-20f, dy = scal[1];
    float den = dx*dx + dy*dy;
    scal[4] = (nx*dx + ny*dy) / den;
    scal[5] = (ny*dx - nx*dy) / den;
    scal[0] = nx; scal[1] = ny;
    if (sqrtf(nx*nx + ny*ny) < 1e-10f) scal[6] = 1.f;
}

__global__ __launch_bounds__(256) void lrm_axpy_zr(float2* __restrict__ z,
        float2* __restrict__ r, const float2* __restrict__ p,
        const float2* __restrict__ Ap, const float* __restrict__ scal)
{
    if (scal[6] > 0.5f) return;
    float ax = scal[2], ay = scal[3];
    int i = blockIdx.x * 256 + threadIdx.x;
    float2 pq = p[i], Av = Ap[i], zv = z[i], rv = r[i];
    z[i] = make_float2(zv.x + ax*pq.x - ay*pq.y, zv.y + ax*pq.y + ay*pq.x);
    r[i] = make_float2(rv.x - (ax*Av.x - ay*Av.y), rv.y - (ax*Av.y + ay*Av.x));
}

__global__ __launch_bounds__(256) void lrm_pupd(float2* __restrict__ p,
        const float2* __restrict__ r, const float* __restrict__ scal)
{
    if (scal[6] > 0.5f) return;
    float bx = scal[4], by = scal[5];
    int i = blockIdx.x * 256 + threadIdx.x;
    float2 pq = p[i], rv = r[i];
    p[i] = make_float2(rv.x + bx*pq.x - by*pq.y, rv.y + bx*pq.y + by*pq.x);
}

// ---------------------------------------------------------------------------
extern "C" void kernel_launch(void* const* d_in, const int* in_sizes, int n_in,
                              void* d_out, int out_size, void* d_ws, size_t ws_size,
                              hipStream_t stream)
{
    const float* kr  = (const float*)d_in[0];
    const float* ki  = (const float*)d_in[1];
    const float* mor = (const float*)d_in[2];
    const float* moi = (const float*)d_in[3];
    const float* sr  = (const float*)d_in[4];
    const float* si  = (const float*)d_in[5];
    const float* tbr = (const float*)d_in[6];
    const float* tbi = (const float*)d_in[7];
    const float* msk = (const float*)d_in[8];

    // pick coil-chunk size from available workspace (deterministic per run)
    const size_t fixed = (size_t)NBT*8 + (size_t)NBSV*3*8 + 1024*8 + 64;
    int nc = 1;
    if      ((size_t)NBT*10*8 + fixed <= ws_size) nc = 10;
    else if ((size_t)NBT*5 *8 + fixed <= ws_size) nc = 5;
    else if ((size_t)NBT*2 *8 + fixed <= ws_size) nc = 2;
    const int nchunks = Cc / nc;

    // workspace: tmp[NBT*nc] comb[NBT] p[NBSV] r[NBSV] Ap[NBSV] part[1024] scal
    float2* tmp  = (float2*)d_ws;
    float2* comb = tmp + (size_t)NBT * nc;
    float2* pvec = comb + NBT;
    float2* rvec = pvec + NBSV;
    float2* Apv  = rvec + NBSV;
    float2* part = Apv + NBSV;
    float*  scal = (float*)(part + 1024);
    float2* zvec = (float2*)d_out;      // [b,sv,h,w,2] == float2[b,sv,h,w]

    dim3 b256(256), b512(512), b1(1);
    dim3 gRowsC(Bb*Tt*nc*Hh/8);   // coil-chunked row pass
    dim3 gRows(Bb*Tt*Hh/8);       // 768 (rowadj)
    dim3 gCol(Bb*Tt*nc*16);
    dim3 gPack(NBT/256*nc);
    dim3 gPix(NBSV/256);          // 1536
    dim3 gDot(NBSV/512);          // 768

    // ---- b_vec = sum_t conj(L) * adjoint(Y) + LAM*MO ; init CG state
    for (int ch = 0; ch < nchunks; ++ch) {
        int c0 = ch * nc;
        hipLaunchKernelGGL(lrm_pack,   gPack, b256, 0, stream, kr, ki, c0, nc, tmp);
        hipLaunchKernelGGL(lrm_col,    gCol,  b512, 0, stream, tmp, (const float*)nullptr, nc, 0);
        hipLaunchKernelGGL(lrm_rowadj, gRows, b256, 0, stream, tmp, sr, si, c0, nc, comb, (ch > 0) ? 1 : 0);
    }
    hipLaunchKernelGGL(lrm_proj_init, gPix, b256, 0, stream, comb, tbr, tbi, mor, moi, rvec, pvec, zvec);
    hipLaunchKernelGGL(lrm_dot,       gDot, b256, 0, stream, rvec, rvec, part);
    hipLaunchKernelGGL(lrm_dot_final, dim3(1), b256, 0, stream, part, scal);
    hipLaunchKernelGGL(lrm_s_init,    dim3(1), b1, 0, stream, scal);

    // ---- 6 CG iterations
    for (int it = 0; it < 6; ++it) {
        for (int ch = 0; ch < nchunks; ++ch) {
            int c0 = ch * nc;
            hipLaunchKernelGGL(lrm_rowfwd, gRowsC, b256, 0, stream, pvec, tbr, tbi, sr, si, c0, nc, tmp);
            hipLaunchKernelGGL(lrm_col,    gCol,   b512, 0, stream, tmp, msk, nc, 1);
            hipLaunchKernelGGL(lrm_rowadj, gRows,  b256, 0, stream, tmp, sr, si, c0, nc, comb, (ch > 0) ? 1 : 0);
        }
        hipLaunchKernelGGL(lrm_proj_ap,   gPix, b256, 0, stream, comb, tbr, tbi, pvec, Apv);
        hipLaunchKernelGGL(lrm_dot,       gDot, b256, 0, stream, pvec, Apv, part);
        hipLaunchKernelGGL(lrm_dot_final, dim3(1), b256, 0, stream, part, scal);
        hipLaunchKernelGGL(lrm_s_alpha,   dim3(1), b1, 0, stream, scal);
        hipLaunchKernelGGL(lrm_axpy_zr,   gPix, b256, 0, stream, zvec, rvec, pvec, Apv, scal);
        hipLaunchKernelGGL(lrm_dot,       gDot, b256, 0, stream, rvec, rvec, part);
        hipLaunchKernelGGL(lrm_dot_final, dim3(1), b256, 0, stream, part, scal);
        hipLaunchKernelGGL(lrm_s_beta,    dim3(1), b1, 0, stream, scal);
        hipLaunchKernelGGL(lrm_pupd,      gPix, b256, 0, stream, pvec, rvec, scal);
    }
    (void)in_sizes; (void)n_in; (void)out_size; (void)ws_size;
}